// SparseNodeConv_89275190215169
// MI455X (gfx1250) — compile-verified
//
#include <hip/hip_runtime.h>
#include <hip/hip_bf16.h>

typedef __attribute__((ext_vector_type(2))) float v2f;
typedef __attribute__((ext_vector_type(8))) float v8f;

#define DFEAT 128

// ---------------------------------------------------------------------------
// Kernel A: edge scatter-add.  out[src] += msgs[dst]  for each edge.
// One wave per edge; each lane moves 4 consecutive floats:
//   global_load_b128 gather from msgs[dst], then 4x hardware
//   global_atomic_add_f32 (no return, device scope) into out[src] via
//   inline CDNA5 asm — guarantees the native fp32 atomic, never a CAS loop.
// ---------------------------------------------------------------------------
__global__ __launch_bounds__(256) void SparseNodeConv_scatter(
    const float* __restrict__ msgs,
    const int* __restrict__ esrc,
    const int* __restrict__ edst,
    float* __restrict__ out,
    int E)
{
    const int edge = (int)((blockIdx.x * (unsigned)blockDim.x + threadIdx.x) >> 5);
    const int lane = threadIdx.x & 31;
    if (edge >= E) return;

    const int s = esrc[edge];   // broadcast load (same addr across wave)
    const int d = edst[edge];

    const float4 v = *(const float4*)(msgs + (size_t)d * DFEAT + lane * 4);
    float* o = out + (size_t)s * DFEAT + lane * 4;

    // Non-returning fp32 atomic adds, device scope (L2 atomic units).
    asm volatile(
        "global_atomic_add_f32 %0, %1, off scope:SCOPE_DEV\n\t"
        "global_atomic_add_f32 %0, %2, off offset:4 scope:SCOPE_DEV\n\t"
        "global_atomic_add_f32 %0, %3, off offset:8 scope:SCOPE_DEV\n\t"
        "global_atomic_add_f32 %0, %4, off offset:12 scope:SCOPE_DEV"
        :
        : "v"(o), "v"(v.x), "v"(v.y), "v"(v.z), "v"(v.w)
        : "memory");
}

// ---------------------------------------------------------------------------
// Kernel B: fused dual GEMM with fp32 WMMA.
//   msgs = node_attr @ node_weight            -> d_ws
//   out  = node_attr @ root + bias            -> d_out (initializes output)
// One workgroup = 16 rows; 8 waves, wave w owns columns [16w, 16w+16).
// K-loop: 32 steps of V_WMMA_F32_16X16X4_F32, two accumulators sharing the
// A fragment.  Epilogue branch is UNIFORM, fast path = clause of 16
// global_store_b32 with 512 B immediate offsets.
// ---------------------------------------------------------------------------
__global__ __launch_bounds__(256) void SparseNodeConv_gemm_wmma(
    const float* __restrict__ A,     // [N, 128] node_attr
    const float* __restrict__ W,     // [128, 128] node_weight
    const float* __restrict__ R,     // [128, 128] root
    const float* __restrict__ bias,  // [128]
    float* __restrict__ msgs,        // [N, 128] workspace
    float* __restrict__ out,         // [N, 128] output
    int N)
{
    const int wave = threadIdx.x >> 5;     // 0..7 -> column block
    const int lane = threadIdx.x & 31;
    const int half = lane >> 4;            // 0 or 1
    const int mn   = lane & 15;            // row (A) / col (B,C,D) within tile

    const int row0 = blockIdx.x * 16;
    const int col0 = wave * 16;

    // Clamp A row for loads so EXEC stays all-ones (WMMA requirement).
    const int rA = min(row0 + mn, N - 1);
    const float* __restrict__ arow = A + (size_t)rA * DFEAT;

    v8f accW = {};  // msgs accumulator
    v8f accR = {};  // root accumulator

    #pragma unroll 4
    for (int k0 = 0; k0 < DFEAT; k0 += 4) {
        const int ka = k0 + 2 * half;      // K index pair handled by this lane

        // A fragment: 16x4 fp32 tile. VGPR0 = A[m][ka], VGPR1 = A[m][ka+1]
        v2f a = *(const v2f*)(arow + ka);  // 8B-aligned (ka is even)

        // B fragments: 4x16 fp32 tiles of W and R (row-major, stride 128)
        const size_t bi0 = (size_t)ka * DFEAT + col0 + mn;
        v2f bw, br;
        bw.x = W[bi0];
        bw.y = W[bi0 + DFEAT];
        br.x = R[bi0];
        br.y = R[bi0 + DFEAT];

        // (neg_a, A, neg_b, B, c_mod, C, reuse_a, reuse_b)
        accW = __builtin_amdgcn_wmma_f32_16x16x4_f32(
            false, a, false, bw, (short)0, accW, false, false);
        accR = __builtin_amdgcn_wmma_f32_16x16x4_f32(
            false, a, false, br, (short)0, accR, false, false);
    }

    const float bv = bias[col0 + mn];

    // C/D layout: VGPR r holds row (r + 8*half), column mn of the 16x16 tile.
    // Lane's 8 elements live at consecutive row stride (512 B immediates).
    const size_t base = (size_t)(row0 + 8 * half) * DFEAT + col0 + mn;
    float* __restrict__ mp = msgs + base;
    float* __restrict__ op = out + base;

    if (row0 + 16 <= N) {               // uniform branch: full tile
        #pragma unroll
        for (int r = 0; r < 8; ++r) {
            mp[(size_t)r * DFEAT] = accW[r];
            op[(size_t)r * DFEAT] = accR[r] + bv;
        }
    } else {                            // uniform tail path (unused for N%16==0)
        #pragma unroll
        for (int r = 0; r < 8; ++r) {
            if (row0 + 8 * half + r < N) {
                mp[(size_t)r * DFEAT] = accW[r];
                op[(size_t)r * DFEAT] = accR[r] + bv;
            }
        }
    }
}

// ---------------------------------------------------------------------------
// Inputs (setup_inputs order):
//   0: node_attr  [N,128] f32      3: node_weight [128,128] f32
//   1: edge_src   [E]     i32      4: root        [128,128] f32
//   2: edge_dst   [E]     i32      5: bias        [128]     f32
// Output: [N,128] f32
// Workspace: msgs [N,128] f32 = 25.6 MB
// ---------------------------------------------------------------------------
extern "C" void kernel_launch(void* const* d_in, const int* in_sizes, int n_in,
                              void* d_out, int out_size, void* d_ws, size_t ws_size,
                              hipStream_t stream) {
    const float* node_attr   = (const float*)d_in[0];
    const int*   edge_src    = (const int*)  d_in[1];
    const int*   edge_dst    = (const int*)  d_in[2];
    const float* node_weight = (const float*)d_in[3];
    const float* root        = (const float*)d_in[4];
    const float* bias        = (const float*)d_in[5];

    const int N = in_sizes[0] / DFEAT;
    const int E = in_sizes[1];

    float* msgs = (float*)d_ws;
    float* out  = (float*)d_out;

    // Phase 1: dual GEMM (msgs -> ws, root_info + bias -> out)
    const int rowBlocks = (N + 15) / 16;
    SparseNodeConv_gemm_wmma<<<rowBlocks, 256, 0, stream>>>(
        node_attr, node_weight, root, bias, msgs, out, N);

    // Phase 2: scatter-add over edges (1 wave per edge, 8 edges per block)
    const int blocks = (E + 7) / 8;
    SparseNodeConv_scatter<<<blocks, 256, 0, stream>>>(
        msgs, edge_src, edge_dst, out, E);
}